// SinkPIT_38156489457951
// MI455X (gfx1250) — compile-verified
//
#include <hip/hip_runtime.h>

typedef __attribute__((ext_vector_type(2))) float v2f;
typedef __attribute__((ext_vector_type(8))) float v8f;

#define B_ 16
#define N_ 8
#define T_ 65536

constexpr int WAVES_PER_BLOCK  = 8;
constexpr int CHUNK_W          = 1024;                                   // T elems per wave
constexpr int STEPS            = CHUNK_W / 4;                            // 256 WMMAs per wave
constexpr int BLOCKS_PER_BATCH = T_ / (WAVES_PER_BLOCK * CHUNK_W);       // 8
constexpr int WS_FLOATS        = B_ * 16 * 16;                           // 4096

// ---------------------------------------------------------------------------
// Kernel 0: zero the per-batch 16x16 accumulators in workspace (graph-safe).
// ---------------------------------------------------------------------------
__global__ void zero_ws_kernel(float* ws, int n) {
    int i = blockIdx.x * blockDim.x + threadIdx.x;
    if (i < n) ws[i] = 0.0f;
}

// ---------------------------------------------------------------------------
// Kernel 1: fused Gram + norms via V_WMMA_F32_16X16X4_F32.
//   A (16x4): rows 0-7 = input rows, rows 8-15 = target rows
//   B (4x16): cols 0-7 = target rows, cols 8-15 = input rows
//   D[i][j] i,j<8 = x_i.y_j ; D[i][8+i] = |x_i|^2 ; D[8+j][j] = |y_j|^2
// ---------------------------------------------------------------------------
__global__ __launch_bounds__(32 * WAVES_PER_BLOCK)
void pairwise_gram_wmma(const float* __restrict__ inp,
                        const float* __restrict__ tgt,
                        float* __restrict__ ws) {
    const int b    = blockIdx.x / BLOCKS_PER_BATCH;
    const int s    = blockIdx.x % BLOCKS_PER_BATCH;
    const int wave = threadIdx.x >> 5;
    const int lane = threadIdx.x & 31;
    const int half = lane >> 4;      // 0: K=0,1   1: K=2,3
    const int lrow = lane & 15;      // M for A, N for B

    const float* inB  = inp + (size_t)b * N_ * T_;
    const float* tgtB = tgt + (size_t)b * N_ * T_;

    // Loop-invariant row base pointers (A-lane row, B-lane column source).
    const float* arow = (lrow < 8) ? (inB  + (size_t)lrow * T_)
                                   : (tgtB + (size_t)(lrow - 8) * T_);
    const float* brow = (lrow < 8) ? (tgtB + (size_t)lrow * T_)
                                   : (inB  + (size_t)(lrow - 8) * T_);

    // Fold the per-wave K-slice start and the lane's K-offset into the base
    // pointers ONCE, so the loop below has a uniform, compile-time trip count
    // (scalar branch, no EXEC masking around the WMMA) and unrolls cleanly.
    const int start = s * (WAVES_PER_BLOCK * CHUNK_W) + wave * CHUNK_W + 2 * half;
    const float* ap = arow + start;
    const float* bp = brow + start;

    v8f acc = {0.f, 0.f, 0.f, 0.f, 0.f, 0.f, 0.f, 0.f};

    #pragma unroll 4
    for (int k = 0; k < STEPS; ++k) {
        v2f a  = *(const v2f*)(ap + 4 * k);   // global_load_b64, imm offset
        v2f bb = *(const v2f*)(bp + 4 * k);
        // 8 args: (neg_a, A, neg_b, B, c_mod, C, reuse_a, reuse_b)
        acc = __builtin_amdgcn_wmma_f32_16x16x4_f32(
            false, a, false, bb, (short)0, acc, false, false);
    }

    // C/D layout: VGPR v -> (M = v + 8*half, N = lrow)
    float* D = ws + b * 256;
    #pragma unroll
    for (int v = 0; v < 8; ++v) {
        const int M = v + 8 * half;
        unsafeAtomicAdd(&D[M * 16 + lrow], acc[v]);   // global_atomic_add_f32
    }
}

// ---------------------------------------------------------------------------
// Kernel 2: finalize. One thread per batch does the 8x8 Sinkhorn (tiny).
// d_out layout: [0] = mean loss ; [1 .. 128] = pattern[b][i] as float.
// ---------------------------------------------------------------------------
__global__ void sinkhorn_finalize(const float* __restrict__ ws,
                                  float* __restrict__ out) {
    __shared__ float lossSh[B_];
    const int b = threadIdx.x;
    if (b < B_) {
        const float* D = ws + b * 256;
        const float invT = 1.0f / (float)T_;
        float pl[8][8], Z[8][8];
        for (int i = 0; i < 8; ++i) {
            const float sx = D[i * 16 + 8 + i];
            for (int j = 0; j < 8; ++j) {
                const float sy = D[(8 + j) * 16 + j];
                const float v  = (sx - 2.0f * D[i * 16 + j] + sy) * invT;
                pl[i][j] = v;
                Z[i][j]  = -v;   // COLDNESS = 1
            }
        }
        for (int it = 0; it < 10; ++it) {
            // axis=1: logsumexp over i, per column j
            for (int j = 0; j < 8; ++j) {
                float m = Z[0][j];
                for (int i = 1; i < 8; ++i) m = fmaxf(m, Z[i][j]);
                float sum = 0.f;
                for (int i = 0; i < 8; ++i) sum += expf(Z[i][j] - m);
                const float l = m + logf(sum);
                for (int i = 0; i < 8; ++i) Z[i][j] -= l;
            }
            // axis=2: logsumexp over j, per row i
            for (int i = 0; i < 8; ++i) {
                float m = Z[i][0];
                for (int j = 1; j < 8; ++j) m = fmaxf(m, Z[i][j]);
                float sum = 0.f;
                for (int j = 0; j < 8; ++j) sum += expf(Z[i][j] - m);
                const float l = m + logf(sum);
                for (int j = 0; j < 8; ++j) Z[i][j] -= l;
            }
        }
        float loss = 0.f;
        for (int i = 0; i < 8; ++i)
            for (int j = 0; j < 8; ++j) {
                const float P = expf(Z[i][j]);
                loss += (pl[i][j] + Z[i][j]) * P;
            }
        lossSh[b] = loss;
        for (int i = 0; i < 8; ++i) {
            int best = 0; float bv = Z[i][0];
            for (int j = 1; j < 8; ++j)
                if (Z[i][j] > bv) { bv = Z[i][j]; best = j; }
            out[1 + b * 8 + i] = (float)best;
        }
    }
    __syncthreads();
    if (threadIdx.x == 0) {
        float s = 0.f;
        for (int i = 0; i < B_; ++i) s += lossSh[i];
        out[0] = s / (float)B_;
    }
}

// ---------------------------------------------------------------------------
extern "C" void kernel_launch(void* const* d_in, const int* in_sizes, int n_in,
                              void* d_out, int out_size, void* d_ws, size_t ws_size,
                              hipStream_t stream) {
    const float* inp = (const float*)d_in[0];
    const float* tgt = (const float*)d_in[1];
    float* ws  = (float*)d_ws;
    float* out = (float*)d_out;

    zero_ws_kernel<<<(WS_FLOATS + 255) / 256, 256, 0, stream>>>(ws, WS_FLOATS);

    pairwise_gram_wmma<<<B_ * BLOCKS_PER_BATCH, 32 * WAVES_PER_BLOCK, 0, stream>>>(
        inp, tgt, ws);

    sinkhorn_finalize<<<1, 32, 0, stream>>>(ws, out);
}